// CompoundWordAutoregressiveWrapper_38912403701970
// MI455X (gfx1250) — compile-verified
//
#include <hip/hip_runtime.h>
#include <stdint.h>

#define VOCAB 6913
#define NV4   1728              // (VOCAB-1)/4 full float4 chunks of the LDS row
#define NPOS  2048
#define NHEAD 7
#define NROWS (NHEAD * NPOS)
#define BLK   256

// ---------------- CDNA5 async global->LDS helpers (inline asm) ----------------
// VDST VGPR = LDS byte address (wave-relative), VADDR = 64-bit global address.
// Low 32 bits of a generic pointer to __shared__ data are exactly the LDS
// byte offset (AS3->generic addrspacecast places the aperture in the high
// word), so a simple truncation gives the VDST value without any AS casts.

__device__ __forceinline__ void async_copy16(const void* g, void* l) {
  uint32_t loff  = (uint32_t)(uintptr_t)l;
  uint64_t gaddr = (uint64_t)(uintptr_t)g;
  asm volatile("global_load_async_to_lds_b128 %0, %1, off"
               :: "v"(loff), "v"(gaddr) : "memory");
}

__device__ __forceinline__ void async_copy4(const void* g, void* l) {
  uint32_t loff  = (uint32_t)(uintptr_t)l;
  uint64_t gaddr = (uint64_t)(uintptr_t)g;
  asm volatile("global_load_async_to_lds_b32 %0, %1, off"
               :: "v"(loff), "v"(gaddr) : "memory");
}

__device__ __forceinline__ void wait_async0() {
#if __has_builtin(__builtin_amdgcn_s_wait_asynccnt)
  __builtin_amdgcn_s_wait_asynccnt(0);
#else
  asm volatile("s_wait_asynccnt 0" ::: "memory");
#endif
}

// ---------------- Kernel 1: per-row masked NLL ----------------
// One block per (head, position). Row of 6913 f32 staged to LDS via async
// copy (whole row in flight -> deep MLP), then max and sum(exp(x-max))
// passes out of LDS with ds_load_b128.
__global__ __launch_bounds__(BLK) void ce_rows_kernel(
    const int* __restrict__ x,
    const float* __restrict__ p0, const float* __restrict__ p1,
    const float* __restrict__ p2, const float* __restrict__ p3,
    const float* __restrict__ p4, const float* __restrict__ p5,
    const float* __restrict__ p6,
    float* __restrict__ row_nll)
{
  __shared__ __align__(16) float row[VOCAB];
  __shared__ float red[12];

  const int tid = threadIdx.x;
  const int rid = blockIdx.x;            // 0 .. NROWS-1
  const int h   = rid >> 11;             // head index 0..6 (NPOS = 2048)
  const int pos = rid & (NPOS - 1);      // 0..2047  (b = pos>>10, s = pos&1023)

  const float* heads[NHEAD] = {p0, p1, p2, p3, p4, p5, p6};
  const float* g = heads[h] + (size_t)pos * VOCAB;

  // Rows are only 4B-aligned (6913 floats). Split: lead scalars (0..3),
  // 16B-aligned b128 body, tail scalars (0..3). Lead elements are parked at
  // the END of the LDS buffer; reductions are order-agnostic.
  const uintptr_t base = (uintptr_t)g;
  const int lead = (int)(((16u - (unsigned)(base & 15u)) & 15u) >> 2);
  const int nv4  = (VOCAB - lead) >> 2;  // 1727 or 1728
  const int tail = (VOCAB - lead) & 3;

  const char* ga = (const char*)(g + lead);    // 16B aligned
  char* lrow = (char*)row;
#pragma unroll
  for (int it = 0; it < 7; ++it) {             // 7*256 >= 1728 >= nv4
    const int i = tid + it * BLK;
    if (i < nv4) async_copy16(ga + 16 * i, lrow + 16 * i);
  }
  if (tid < lead)                              // elements [0, lead)
    async_copy4(g + tid, &row[VOCAB - lead + tid]);
  if (tid < tail) {                            // elements [lead+4*nv4, VOCAB)
    const int e = lead + 4 * nv4 + tid;
    async_copy4(g + e, &row[e - lead]);
  }
  wait_async0();
  __syncthreads();

  const float4* row4 = (const float4*)row;     // 1728 vec4 + row[6912]

  // ---- pass 1: row max ----
  float m = -__builtin_inff();
#pragma unroll
  for (int it = 0; it < 7; ++it) {
    const int i = tid + it * BLK;
    if (i < NV4) {
      const float4 v = row4[i];
      m = fmaxf(m, fmaxf(fmaxf(v.x, v.y), fmaxf(v.z, v.w)));
    }
  }
  if (tid == 0) m = fmaxf(m, row[VOCAB - 1]);
  for (int off = 16; off > 0; off >>= 1) m = fmaxf(m, __shfl_xor(m, off));
  if ((tid & 31) == 0) red[tid >> 5] = m;
  __syncthreads();
  if (tid == 0) {
    float mm = red[0];
    for (int w = 1; w < BLK / 32; ++w) mm = fmaxf(mm, red[w]);
    red[8] = mm;
  }
  __syncthreads();
  m = red[8];

  // ---- pass 2: sum exp(x - max) ----
  float ssum = 0.0f;
#pragma unroll
  for (int it = 0; it < 7; ++it) {
    const int i = tid + it * BLK;
    if (i < NV4) {
      const float4 v = row4[i];
      ssum += __expf(v.x - m) + __expf(v.y - m) +
              __expf(v.z - m) + __expf(v.w - m);
    }
  }
  if (tid == 0) ssum += __expf(row[VOCAB - 1] - m);
  for (int off = 16; off > 0; off >>= 1) ssum += __shfl_xor(ssum, off);
  __syncthreads();                       // red[] reuse guard
  if ((tid & 31) == 0) red[tid >> 5] = ssum;
  __syncthreads();

  if (tid == 0) {
    float s = 0.0f;
    for (int w = 0; w < BLK / 32; ++w) s += red[w];
    const int b = pos >> 10, sp = pos & 1023;
    const int* xb = x + ((b * 1025) + sp + 1) * 12;   // target = x[:,1:,:]
    const int tgt   = xb[h];
    const int maskv = xb[0];
    const int slot  = (tgt >= lead) ? (tgt - lead) : (VOCAB - lead + tgt);
    const float xt  = row[slot];
    const float nll = -(xt - m - __logf(s));
    row_nll[rid] = (maskv != 0) ? nll : 0.0f;
  }
}

// ---------------- Kernel 2: finalize 8 outputs ----------------
__global__ __launch_bounds__(BLK) void finalize_kernel(
    const int* __restrict__ x, const float* __restrict__ row_nll,
    float* __restrict__ out)
{
  __shared__ float red[8];
  __shared__ float tot[9];
  const int tid = threadIdx.x;

  // f0 = constant value of f[...,0]: 576 * sum_{i=0..11} sin(-pi*i/6)  (~0)
  float f0 = 0.0f;
  for (int i = 0; i < 12; ++i) f0 += __sinf((float)i * (-0.52359877559829887f));
  f0 *= 576.0f;

  float acc[9];
  for (int k = 0; k < 9; ++k) acc[k] = 0.0f;

#pragma unroll
  for (int it = 0; it < NPOS / BLK; ++it) {    // 8 compile-time iterations
    const int pos = tid + it * BLK;
    const int b = pos >> 10, sp = pos & 1023;
    const int* xb = x + ((b * 1025) + sp + 1) * 12;
    const float mk = (xb[0] != 0) ? 1.0f : 0.0f;
    acc[0] += mk;
#pragma unroll
    for (int c = 0; c < NHEAD; ++c)
      acc[1 + c] += row_nll[c * NPOS + pos];   // already masked
    const float dlt = f0 - (float)xb[11];
    acc[8] += mk * dlt * dlt;
  }

  for (int k = 0; k < 9; ++k) {
    float v = acc[k];
    for (int off = 16; off > 0; off >>= 1) v += __shfl_xor(v, off);
    if ((tid & 31) == 0) red[tid >> 5] = v;
    __syncthreads();
    if (tid == 0) {
      float t = 0.0f;
      for (int w = 0; w < BLK / 32; ++w) t += red[w];
      tot[k] = t;
    }
    __syncthreads();
  }

  if (tid == 0) {
    const float cnt = tot[0];
    for (int c = 0; c < NHEAD; ++c)
      out[c] = (cnt == 0.0f) ? 0.0f : tot[1 + c] / cnt;
    out[7] = (cnt == 0.0f) ? 0.0f : tot[8] / cnt;
  }
}

extern "C" void kernel_launch(void* const* d_in, const int* in_sizes, int n_in,
                              void* d_out, int out_size, void* d_ws, size_t ws_size,
                              hipStream_t stream) {
  const int*   x = (const int*)d_in[0];
  const float* p0 = (const float*)d_in[1];
  const float* p1 = (const float*)d_in[2];
  const float* p2 = (const float*)d_in[3];
  const float* p3 = (const float*)d_in[4];
  const float* p4 = (const float*)d_in[5];
  const float* p5 = (const float*)d_in[6];
  const float* p6 = (const float*)d_in[7];
  float* row_nll = (float*)d_ws;                 // NROWS * 4 bytes

  ce_rows_kernel<<<NROWS, BLK, 0, stream>>>(x, p0, p1, p2, p3, p4, p5, p6, row_nll);
  finalize_kernel<<<1, BLK, 0, stream>>>(x, row_nll, (float*)d_out);
}